// SINTokInputEmb_52295521796611
// MI455X (gfx1250) — compile-verified
//
#include <hip/hip_runtime.h>
#include <math.h>

// ---------------------------------------------------------------------------
// SINTokInputEmb for MI455X (gfx1250, wave32)
//   out = LayerNorm( word_emb[ids] + PE[pos] + type_emb[tt] +
//                    PE[s0] + PE[s1] + PE[s2] ) * gamma + beta
// Memory-bound gather+LN: no matmul -> no WMMA. We use the CDNA5 async-to-LDS
// path (ASYNCcnt) to double-buffer the random word-row gather, global_prefetch
// for PE rows, and nontemporal stores so the 100MB output stream does not
// evict the 154MB embedding table from the 192MB L2.
// ---------------------------------------------------------------------------

typedef float f4 __attribute__((ext_vector_type(4)));

#define TOKENS          32768      // B*N = 8*4096
#define N_SEQ           4096
#define HID             768
#define F4_PER_ROW      192        // 768/4
#define MAXPOS          5000
#define LN_EPS          1e-12f
#define WAVES_PER_BLOCK 8
#define TOK_PER_WAVE    4
// freq(k) = exp(-ln(10000) * k / 384), k = pair index 0..383
#define CFREQ           (-0.0239852614f)

// ---- CDNA5 async copy: one 768-float row, 6 x b128 per wave (ASYNCcnt) ----
__device__ __forceinline__ void async_row_to_lds(const float* grow, unsigned lds_base) {
    const int lane = (int)(threadIdx.x & 31u);
    unsigned long long ga = (unsigned long long)grow + (unsigned long long)(lane * 16);
    unsigned loff = lds_base + (unsigned)(lane * 16);
    // INST_OFFSET is added to BOTH the global and LDS address (ISA 08 §4.4)
    asm volatile("global_load_async_to_lds_b128 %0, %1, off"             :: "v"(loff), "v"(ga) : "memory");
    asm volatile("global_load_async_to_lds_b128 %0, %1, off offset:512"  :: "v"(loff), "v"(ga) : "memory");
    asm volatile("global_load_async_to_lds_b128 %0, %1, off offset:1024" :: "v"(loff), "v"(ga) : "memory");
    asm volatile("global_load_async_to_lds_b128 %0, %1, off offset:1536" :: "v"(loff), "v"(ga) : "memory");
    asm volatile("global_load_async_to_lds_b128 %0, %1, off offset:2048" :: "v"(loff), "v"(ga) : "memory");
    asm volatile("global_load_async_to_lds_b128 %0, %1, off offset:2560" :: "v"(loff), "v"(ga) : "memory");
}

// ---- build PE table into workspace (5000 x 768 fp32 = 15.36 MB, L2-resident)
__global__ __launch_bounds__(256) void pe_build_kernel(float* __restrict__ pe) {
    const int idx = (int)(blockIdx.x * 256 + threadIdx.x);   // pair index
    if (idx >= MAXPOS * (HID / 2)) return;
    const int p = idx / (HID / 2);
    const int k = idx - p * (HID / 2);
    const float f = __expf((float)k * CFREQ);
    float s, c;
    __sincosf((float)p * f, &s, &c);
    pe[(size_t)p * HID + 2 * k]     = s;
    pe[(size_t)p * HID + 2 * k + 1] = c;
}

// ---- fused gather + add + LayerNorm; one wave per token -------------------
template <bool USE_TABLE>
__global__ __launch_bounds__(256) void emb_ln_kernel(
    const int*   __restrict__ input_ids,
    const int*   __restrict__ struct_vec,   // (T,3)
    const int*   __restrict__ token_type,
    const float* __restrict__ word_emb,     // (50257,768)
    const float* __restrict__ type_emb,     // (2,768)
    const float* __restrict__ gamma,
    const float* __restrict__ beta,
    const float* __restrict__ pe,           // (5000,768) in ws (table path)
    float*       __restrict__ out)
{
    __shared__ f4 smem[WAVES_PER_BLOCK][2][F4_PER_ROW];   // 48 KB / block

    const int w    = (int)(threadIdx.x >> 5);
    const int lane = (int)(threadIdx.x & 31u);
    const int t0   = ((int)blockIdx.x * WAVES_PER_BLOCK + w) * TOK_PER_WAVE;

    unsigned lds_off[2];
    lds_off[0] = (unsigned)(unsigned long long)&smem[w][0][0];  // low 32 bits == LDS byte addr
    lds_off[1] = (unsigned)(unsigned long long)&smem[w][1][0];

    // per-lane frequencies for the on-the-fly PE fallback
    float fr0[6], fr1[6];
    if (!USE_TABLE) {
#pragma unroll
        for (int c = 0; c < 6; ++c) {
            const int j = c * 32 + lane;          // f4 index -> pair indices 2j, 2j+1
            fr0[c] = __expf((float)(2 * j)     * CFREQ);
            fr1[c] = __expf((float)(2 * j + 1) * CFREQ);
        }
    }

    // kick off token 0's word row
    {
        const int id0 = __builtin_amdgcn_readfirstlane(input_ids[t0]);
        async_row_to_lds(word_emb + (size_t)id0 * HID, lds_off[0]);
    }

    const f4* gam4 = (const f4*)gamma;
    const f4* bet4 = (const f4*)beta;

#pragma unroll
    for (int i = 0; i < TOK_PER_WAVE; ++i) {
        const int t   = t0 + i;
        const int tt  = __builtin_amdgcn_readfirstlane(token_type[t]);
        const int s0  = __builtin_amdgcn_readfirstlane(struct_vec[3 * t + 0]);
        const int s1  = __builtin_amdgcn_readfirstlane(struct_vec[3 * t + 1]);
        const int s2  = __builtin_amdgcn_readfirstlane(struct_vec[3 * t + 2]);
        const int pos = t & (N_SEQ - 1);

        if (i + 1 < TOK_PER_WAVE) {
            // issue next token's word row into the other LDS buffer
            const int idn = __builtin_amdgcn_readfirstlane(input_ids[t + 1]);
            async_row_to_lds(word_emb + (size_t)idn * HID, lds_off[(i + 1) & 1]);
            if (USE_TABLE) {
                // prefetch next token's gathered PE rows (3 KB each, 24 lanes x 128B)
                const int n0 = __builtin_amdgcn_readfirstlane(struct_vec[3 * (t + 1) + 0]);
                const int n1 = __builtin_amdgcn_readfirstlane(struct_vec[3 * (t + 1) + 1]);
                const int n2 = __builtin_amdgcn_readfirstlane(struct_vec[3 * (t + 1) + 2]);
                if (lane < 24) {
                    __builtin_prefetch(pe + (size_t)n0 * HID + lane * 32, 0, 3);
                    __builtin_prefetch(pe + (size_t)n1 * HID + lane * 32, 0, 3);
                    __builtin_prefetch(pe + (size_t)n2 * HID + lane * 32, 0, 3);
                }
            }
            // 6 newer ops may stay in flight; wait for current buffer's 6
            asm volatile("s_wait_asynccnt 6" ::: "memory");
        } else {
            asm volatile("s_wait_asynccnt 0" ::: "memory");
        }

        const f4* sbuf = &smem[w][i & 1][0];
        const f4* trow = (const f4*)(type_emb + (size_t)tt * HID);
        const f4 *p0 = nullptr, *p1 = nullptr, *p2 = nullptr, *p3 = nullptr;
        if (USE_TABLE) {
            p0 = (const f4*)(pe + (size_t)pos * HID);
            p1 = (const f4*)(pe + (size_t)s0  * HID);
            p2 = (const f4*)(pe + (size_t)s1  * HID);
            p3 = (const f4*)(pe + (size_t)s2  * HID);
        }

        f4 emb[6];
        float sum = 0.f, sumsq = 0.f;
#pragma unroll
        for (int c = 0; c < 6; ++c) {
            const int j = c * 32 + lane;
            f4 e = sbuf[j];            // word row from LDS (async-filled)
            e += trow[j];
            if (USE_TABLE) {
                e += p0[j]; e += p1[j]; e += p2[j]; e += p3[j];
            } else {
                f4 pacc = {0.f, 0.f, 0.f, 0.f};
                const float f0 = fr0[c], f1 = fr1[c];
                const int pp[4] = {pos, s0, s1, s2};
#pragma unroll
                for (int q = 0; q < 4; ++q) {
                    float sa, ca, sb, cb;
                    __sincosf((float)pp[q] * f0, &sa, &ca);
                    __sincosf((float)pp[q] * f1, &sb, &cb);
                    pacc.x += sa; pacc.y += ca; pacc.z += sb; pacc.w += cb;
                }
                e += pacc;
            }
            emb[c] = e;
            sum   += e.x + e.y + e.z + e.w;
            sumsq += e.x * e.x + e.y * e.y + e.z * e.z + e.w * e.w;
        }

        // wave32 reduction (no LDS, no barrier)
#pragma unroll
        for (int off = 16; off >= 1; off >>= 1) {
            sum   += __shfl_xor(sum,   off, 32);
            sumsq += __shfl_xor(sumsq, off, 32);
        }
        const float inv  = 1.0f / (float)HID;
        const float mean = sum * inv;
        const float var  = sumsq * inv - mean * mean;
        const float rstd = rsqrtf(var + LN_EPS);

        f4* orow = (f4*)(out + (size_t)t * HID);
#pragma unroll
        for (int c = 0; c < 6; ++c) {
            const int j = c * 32 + lane;
            f4 o = (emb[c] - mean) * rstd * gam4[j] + bet4[j];
            __builtin_nontemporal_store(o, orow + j);   // keep L2 for the tables
        }
    }
}

extern "C" void kernel_launch(void* const* d_in, const int* in_sizes, int n_in,
                              void* d_out, int out_size, void* d_ws, size_t ws_size,
                              hipStream_t stream) {
    (void)in_sizes; (void)n_in; (void)out_size;
    const int*   ids   = (const int*)d_in[0];
    const int*   sv    = (const int*)d_in[1];
    const int*   tts   = (const int*)d_in[2];
    const float* wemb  = (const float*)d_in[3];
    const float* temb  = (const float*)d_in[4];
    const float* gamma = (const float*)d_in[5];
    const float* beta  = (const float*)d_in[6];
    float*       out   = (float*)d_out;

    const size_t pe_bytes = (size_t)MAXPOS * HID * sizeof(float);   // 15.36 MB
    const int grid = TOKENS / (WAVES_PER_BLOCK * TOK_PER_WAVE);     // 1024

    if (ws_size >= pe_bytes) {
        float* pe = (float*)d_ws;
        const int pairs = MAXPOS * (HID / 2);
        pe_build_kernel<<<(pairs + 255) / 256, 256, 0, stream>>>(pe);
        emb_ln_kernel<true><<<grid, 256, 0, stream>>>(ids, sv, tts, wemb, temb,
                                                      gamma, beta, pe, out);
    } else {
        emb_ln_kernel<false><<<grid, 256, 0, stream>>>(ids, sv, tts, wemb, temb,
                                                       gamma, beta, nullptr, out);
    }
}